// Involution2d_1382979469559
// MI455X (gfx1250) — compile-verified
//
#include <hip/hip_runtime.h>
#include <hip/hip_bf16.h>

typedef __attribute__((ext_vector_type(2))) float v2f;
typedef __attribute__((ext_vector_type(8))) float v8f;

#define C_DIM 288
#define N_PB  4096      // H*W per batch
#define B_DIM 8
#define NTOT  32768.0f  // B*H*W
#define KSTEP 8
#define NSTAGE (C_DIM / KSTEP)   // 36

// Async global -> LDS copy (CDNA5 ASYNCcnt path), 16B per lane.
// Operands: LDS byte address VGPR (VDST field), 64-bit global address VGPR pair.
#define ASYNC_B128(ldsaddr, gaddr)                                          \
    asm volatile("global_load_async_to_lds_b128 %0, %1, off"                \
                 :: "v"(ldsaddr), "v"(gaddr) : "memory")

// ---------------------------------------------------------------------------
// GEMM: Y[b] (288 x 4096) = W (288x288) @ X[b] (288x4096) + bias, optional
// ReLU + per-channel sum/sumsq accumulation (for BatchNorm batch stats).
//
// Block: 8 waves = 32(M) x 256(N) tile. Wave tile: M=16, N=64 (4 v8f accums),
// V_WMMA_F32_16X16X4_F32. The X tile (8 x 256 per stage) is double-buffered
// in LDS, filled with global_load_async_to_lds_b128 (2 ops/thread/stage) and
// consumed via ds_load by both M-wave-rows (halves global X traffic).
// Async loads complete in order: after issuing stage s+1 (2 ops),
// s_wait_asynccnt 0x2 guarantees stage s has landed; barriers publish LDS.
// ---------------------------------------------------------------------------
template <bool RELU_STATS>
__global__ __launch_bounds__(256) void gemm288(
    const float* __restrict__ W,     // 288x288 row-major (out, in)
    const float* __restrict__ X,     // (B, 288, 4096)
    const float* __restrict__ bias,  // 288
    float* __restrict__ Y,           // (B, 288, 4096)
    float* __restrict__ sums,        // 288 (atomic accum)
    float* __restrict__ sumsq)       // 288 (atomic accum)
{
    __shared__ float tileX[2][KSTEP][256];

    const int wave = threadIdx.x >> 5;
    const int lane = threadIdx.x & 31;
    const int wm   = wave >> 2;            // 0..1
    const int wn   = wave & 3;             // 0..3
    const int m0   = blockIdx.y * 32 + wm * 16;
    const int n0   = blockIdx.x * 256 + wn * 64;
    const int half = lane >> 4;            // 0: lanes 0-15, 1: lanes 16-31
    const int lr   = lane & 15;

    const float* __restrict__ Xb = X + (size_t)blockIdx.z * C_DIM * N_PB;
    float* __restrict__       Yb = Y + (size_t)blockIdx.z * C_DIM * N_PB;
    const float* __restrict__ Wrow = W + (size_t)(m0 + lr) * C_DIM;

    // Per-thread async-copy addresses: thread t moves rows (t>>6) and
    // (t>>6)+4 of the stage, columns (t&63)*4 .. +3  (16B contiguous).
    const uint32_t ldsbase =
        (uint32_t)(uintptr_t)&tileX[0][0][0] + threadIdx.x * 16u;
    const uint64_t g0 =
        (uint64_t)(uintptr_t)(Xb + (size_t)(threadIdx.x >> 6) * N_PB +
                              blockIdx.x * 256 + (threadIdx.x & 63) * 4);
    const uint64_t grow4 = (uint64_t)4 * N_PB * sizeof(float);

    // Prologue: stage 0 into buffer 0.
    ASYNC_B128(ldsbase, g0);
    ASYNC_B128(ldsbase + 4096u, g0 + grow4);

    v8f acc0 = {}, acc1 = {}, acc2 = {}, acc3 = {};
    const int ncol = wn * 64 + lr;

    for (int ks = 0; ks < NSTAGE; ++ks) {
        const int buf = ks & 1;
        if (ks + 1 < NSTAGE) {
            const uint32_t lb = ldsbase + (uint32_t)(((ks + 1) & 1) * 8192);
            const uint64_t gb =
                g0 + (uint64_t)(ks + 1) * (KSTEP * N_PB * sizeof(float));
            ASYNC_B128(lb, gb);
            ASYNC_B128(lb + 4096u, gb + grow4);
            asm volatile("s_wait_asynccnt 0x2" ::: "memory"); // stage ks done
        } else {
            asm volatile("s_wait_asynccnt 0x0" ::: "memory");
        }
        __syncthreads();   // all waves' stage-ks LDS writes visible

#pragma unroll
        for (int inner = 0; inner < 2; ++inner) {
            const int kk = inner * 4 + 2 * half;        // LDS row of frag .x
            const int kg = ks * KSTEP + kk;             // global k for A frag
            v2f a;
            a.x = Wrow[kg];
            a.y = Wrow[kg + 1];

            const float* lp = &tileX[buf][kk][ncol];
            v2f b0, b1, b2, b3;
            b0.x = lp[0];   b0.y = lp[256];
            b1.x = lp[16];  b1.y = lp[256 + 16];
            b2.x = lp[32];  b2.y = lp[256 + 32];
            b3.x = lp[48];  b3.y = lp[256 + 48];

            acc0 = __builtin_amdgcn_wmma_f32_16x16x4_f32(false, a, false, b0, (short)0, acc0, false, false);
            acc1 = __builtin_amdgcn_wmma_f32_16x16x4_f32(false, a, false, b1, (short)0, acc1, false, false);
            acc2 = __builtin_amdgcn_wmma_f32_16x16x4_f32(false, a, false, b2, (short)0, acc2, false, false);
            acc3 = __builtin_amdgcn_wmma_f32_16x16x4_f32(false, a, false, b3, (short)0, acc3, false, false);
        }
        __syncthreads();   // protect buf before it is refilled (stage ks+2)
    }

    // Epilogue: bias (+ReLU + stats), store. Lane holds rows m0+half*8+r at
    // column n0 + j*16 + lr.
    const int mbase = m0 + half * 8;
    float ls[8], ls2[8];
#pragma unroll
    for (int r = 0; r < 8; ++r) { ls[r] = 0.f; ls2[r] = 0.f; }

#pragma unroll
    for (int r = 0; r < 8; ++r) {
        const float bv = bias[mbase + r];
        float* yrow = Yb + (size_t)(mbase + r) * N_PB + n0 + lr;
        float v0 = acc0[r] + bv;
        float v1 = acc1[r] + bv;
        float v2 = acc2[r] + bv;
        float v3 = acc3[r] + bv;
        if (RELU_STATS) {
            v0 = fmaxf(v0, 0.f); v1 = fmaxf(v1, 0.f);
            v2 = fmaxf(v2, 0.f); v3 = fmaxf(v3, 0.f);
            ls[r]  += v0 + v1 + v2 + v3;
            ls2[r] += v0 * v0 + v1 * v1 + v2 * v2 + v3 * v3;
        }
        yrow[0]  = v0;
        yrow[16] = v1;
        yrow[32] = v2;
        yrow[48] = v3;
    }

    if (RELU_STATS) {
        // Butterfly sum across the 16 lanes of each wave32 half
        // (xor masks 1,2,4,8 stay within a half).
#pragma unroll
        for (int r = 0; r < 8; ++r) {
            float s  = ls[r];
            float s2 = ls2[r];
#pragma unroll
            for (int m = 1; m < 16; m <<= 1) {
                s  += __shfl_xor(s,  m, 32);
                s2 += __shfl_xor(s2, m, 32);
            }
            if (lr == 0) {
                atomicAdd(&sums[mbase + r], s);
                atomicAdd(&sumsq[mbase + r], s2);
            }
        }
    }
}

// ---------------------------------------------------------------------------
// BN stat finalize: s = gamma * rsqrt(var+eps), t = beta - mean*s
// ---------------------------------------------------------------------------
__global__ void finalize_stats(const float* __restrict__ sums,
                               const float* __restrict__ sumsq,
                               const float* __restrict__ gamma,
                               const float* __restrict__ beta,
                               float* __restrict__ s, float* __restrict__ t)
{
    int c = blockIdx.x * blockDim.x + threadIdx.x;
    if (c < C_DIM) {
        const float inv  = 1.0f / NTOT;
        float mean = sums[c] * inv;
        float var  = sumsq[c] * inv - mean * mean;
        float rstd = rsqrtf(var + 1e-5f);
        float sc   = gamma[c] * rstd;
        s[c] = sc;
        t[c] = beta[c] - mean * sc;
    }
}

// ---------------------------------------------------------------------------
// Fold BN affine into second conv: Wp[o,c] = Wo[o,c]*s[c];
// bp[o] = bo[o] + sum_c Wo[o,c]*t[c]
// ---------------------------------------------------------------------------
__global__ __launch_bounds__(96) void fold_W(
    const float* __restrict__ Wo, const float* __restrict__ bo,
    const float* __restrict__ s, const float* __restrict__ t,
    float* __restrict__ Wp, float* __restrict__ bp)
{
    __shared__ float red[96];
    const int o   = blockIdx.x;
    const int tid = threadIdx.x;
    float p = 0.f;
    for (int c = tid; c < C_DIM; c += 96) {
        float w = Wo[(size_t)o * C_DIM + c];
        Wp[(size_t)o * C_DIM + c] = w * s[c];
        p += w * t[c];
    }
    red[tid] = p;
    __syncthreads();
    if (tid == 0) {
        float tot = bo[o];
        for (int i = 0; i < 96; ++i) tot += red[i];
        bp[o] = tot;
    }
}

// ---------------------------------------------------------------------------
// Involution: out[b, g*9+c, h, w] = sum_k ker[b, g*9+k, h, w] *
//             x[b, g*9+c, h + k/3 - 1, w + k%3 - 1]   (zero pad)
// One thread per (b,g,h,w); 9 ker taps loaded once, 9 channel outputs emitted.
// ---------------------------------------------------------------------------
__global__ __launch_bounds__(256) void involution(
    const float* __restrict__ x, const float* __restrict__ ker,
    float* __restrict__ out)
{
    const int idx = blockIdx.x * blockDim.x + threadIdx.x; // 8*32*4096 total
    const int w = idx & 63;
    const int h = (idx >> 6) & 63;
    const int g = (idx >> 12) & 31;
    const int b = idx >> 17;

    const size_t base = ((size_t)b * C_DIM + g * 9) * N_PB + h * 64 + w;

    float kv[9];
#pragma unroll
    for (int k = 0; k < 9; ++k) kv[k] = ker[base + (size_t)k * N_PB];

#pragma unroll
    for (int c = 0; c < 9; ++c) {
        const float* __restrict__ xc =
            x + ((size_t)b * C_DIM + g * 9 + c) * N_PB;
        float acc = 0.f;
#pragma unroll
        for (int k = 0; k < 9; ++k) {
            const int hh = h + k / 3 - 1;
            const int ww = w + k % 3 - 1;
            float xv = (hh >= 0 && hh < 64 && ww >= 0 && ww < 64)
                           ? xc[hh * 64 + ww] : 0.f;
            acc += kv[k] * xv;
        }
        out[base + (size_t)c * N_PB] = acc;
    }
}

// ---------------------------------------------------------------------------
extern "C" void kernel_launch(void* const* d_in, const int* in_sizes, int n_in,
                              void* d_out, int out_size, void* d_ws, size_t ws_size,
                              hipStream_t stream) {
    const float* x     = (const float*)d_in[0];
    const float* Wi    = (const float*)d_in[1];
    const float* bi    = (const float*)d_in[2];
    const float* gamma = (const float*)d_in[3];
    const float* beta  = (const float*)d_in[4];
    const float* Wo    = (const float*)d_in[5];
    const float* bo    = (const float*)d_in[6];
    float* out = (float*)d_out;
    float* ws  = (float*)d_ws;

    const size_t TENS = (size_t)B_DIM * C_DIM * N_PB;   // 9,437,184 floats
    float* tmp   = ws;                 // (B,288,4096)
    float* ker   = ws + TENS;          // (B,288,4096)
    float* stats = ws + 2 * TENS;      // sums[288], sumsq[288], ...
    float* sums  = stats;
    float* sumsq = stats + C_DIM;
    float* svec  = stats + 2 * C_DIM;
    float* tvec  = stats + 3 * C_DIM;
    float* Wp    = stats + 4 * C_DIM;  // 288*288
    float* bp    = Wp + C_DIM * C_DIM;

    // Atomically-accumulated stats must start at zero every call.
    hipMemsetAsync(stats, 0, 2 * C_DIM * sizeof(float), stream);

    dim3 ggrid(16, 9, B_DIM);   // N:16*256=4096, M:9*32=288, per-batch
    gemm288<true><<<ggrid, 256, 0, stream>>>(Wi, x, bi, tmp, sums, sumsq);

    finalize_stats<<<1, 288, 0, stream>>>(sums, sumsq, gamma, beta, svec, tvec);
    fold_W<<<C_DIM, 96, 0, stream>>>(Wo, bo, svec, tvec, Wp, bp);

    gemm288<false><<<ggrid, 256, 0, stream>>>(Wp, tmp, bp, ker, sums, sumsq);

    involution<<<(B_DIM * 32 * N_PB) / 256, 256, 0, stream>>>(x, ker, out);
}